// UMM_72653666779314
// MI455X (gfx1250) — compile-verified
//
#include <hip/hip_runtime.h>
#include <math.h>

typedef __attribute__((ext_vector_type(2))) float v2f;
typedef __attribute__((ext_vector_type(4))) float v4f;
typedef __attribute__((ext_vector_type(8))) float v8f;

#define KBOX 64
#define BASEC 0.0024787522f /* e^-6, the reference's BASE constant (linear, not log-space!) */

// workspace float layout:
//  [0   .. 255] A0 table: box k -> {-1, lo0+hi0, -(lo0*hi0), 0}   (dim-0 quadratic coeffs)
//  [256 .. 511] A1 table: same for dim 1
//  [512 .. 575] delta[k] = exp(log_uniform[k]+log_pk[k]) - exp(BASE+log_pk[k])
//  [576]        S = sum_k exp(BASE + log_pk[k])

__global__ void umm_setup_kernel(const float* __restrict__ weights,
                                 const float* __restrict__ centers,
                                 const float* __restrict__ log_sizes,
                                 float* __restrict__ ws) {
    __shared__ float red[KBOX];
    const int k = threadIdx.x;            // 64 threads, k = box index
    float w = weights[k];

    // log_softmax(weights): stable max + sum reductions in LDS
    red[k] = w; __syncthreads();
    for (int off = 32; off > 0; off >>= 1) {
        if (k < off) red[k] = fmaxf(red[k], red[k + off]);
        __syncthreads();
    }
    float m = red[0]; __syncthreads();
    red[k] = __expf(w - m); __syncthreads();
    for (int off = 32; off > 0; off >>= 1) {
        if (k < off) red[k] += red[k + off];
        __syncthreads();
    }
    float lse = m + __logf(red[0]); __syncthreads();
    const float log_pk = w - lse;

    const float ls0 = log_sizes[2 * k + 0], ls1 = log_sizes[2 * k + 1];
    const float s0 = __expf(ls0), s1 = __expf(ls1);
    const float c0 = centers[2 * k + 0], c1 = centers[2 * k + 1];
    const float lo0 = c0 - 0.5f * s0, hi0 = c0 + 0.5f * s0;
    const float lo1 = c1 - 0.5f * s1, hi1 = c1 + 0.5f * s1;
    const float log_uniform = -(ls0 + ls1);

    const float vout = __expf(BASEC + log_pk);
    const float vin  = __expf(log_uniform + log_pk);

    // quadratic form coefficients: (x-lo)(hi-x) = [-1, lo+hi, -lo*hi, 0] . [x^2,x,1,0]
    ws[4 * k + 0] = -1.0f;
    ws[4 * k + 1] = lo0 + hi0;
    ws[4 * k + 2] = -(lo0 * hi0);
    ws[4 * k + 3] = 0.0f;
    ws[256 + 4 * k + 0] = -1.0f;
    ws[256 + 4 * k + 1] = lo1 + hi1;
    ws[256 + 4 * k + 2] = -(lo1 * hi1);
    ws[256 + 4 * k + 3] = 0.0f;
    ws[512 + k] = vin - vout;

    red[k] = vout; __syncthreads();
    for (int off = 32; off > 0; off >>= 1) {
        if (k < off) red[k] += red[k + off];
        __syncthreads();
    }
    if (k == 0) ws[576] = red[0];
}

__global__ __launch_bounds__(256)
void umm_main_kernel(const float* __restrict__ X,
                     const float* __restrict__ ws,
                     float* __restrict__ out, int N) {
    const int lane = threadIdx.x & 31;
    const int col  = lane & 15;           // point column (C N-dim) / box row loader (A M-dim)
    const int half = lane >> 4;           // K rows {0,1} vs {2,3} for A and B operands
    const int wid  = blockIdx.x * (blockDim.x >> 5) + (threadIdx.x >> 5);
    const int nWav = gridDim.x * (blockDim.x >> 5);
    const int nStrips = (N + 15) >> 4;

    // Loop-invariant A operands: 4 tiles of 16 boxes, per dim.
    // A(M=box,K): lane l -> M = l&15 (box 16t+col), VGPR j -> K = j + 2*half.
    v2f A0[4], A1[4];
    float dlt[4][8];     // deltas for the 8 C-rows this lane owns: box = 16t + j + 8*half
#pragma unroll
    for (int t = 0; t < 4; ++t) {
        const int n = t * 16 + col;
        const float* a0 = ws + 4 * n + 2 * half;
        const float* a1 = ws + 256 + 4 * n + 2 * half;
        v2f v0; v0.x = a0[0]; v0.y = a0[1]; A0[t] = v0;
        v2f v1; v1.x = a1[0]; v1.y = a1[1]; A1[t] = v1;
        const float* dp = ws + 512 + 16 * t + 8 * half;
        v4f d0 = *(const v4f*)(dp);
        v4f d1 = *(const v4f*)(dp + 4);
        dlt[t][0] = d0.x; dlt[t][1] = d0.y; dlt[t][2] = d0.z; dlt[t][3] = d0.w;
        dlt[t][4] = d1.x; dlt[t][5] = d1.y; dlt[t][6] = d1.z; dlt[t][7] = d1.w;
    }
    const float S = ws[576];

    for (int strip = wid; strip < nStrips; strip += nWav) {
        const int p16 = strip << 4;
        const int pIdx = min(p16 + col, N - 1);   // clamp only matters in tail strip

        const float x0 = X[2 * pIdx + 0];
        const float x1 = X[2 * pIdx + 1];
        __builtin_prefetch(X + 2 * (size_t)(p16 + (nWav << 4)), 0, 1);

        // B(K, N=point): lanes 0-15 hold K={0,1} -> {x^2, x}; lanes 16-31 K={2,3} -> {1, 0}
        v2f B0, B1;
        B0.x = half ? 1.0f : x0 * x0;  B0.y = half ? 0.0f : x0;
        B1.x = half ? 1.0f : x1 * x1;  B1.y = half ? 0.0f : x1;

        float acc = 0.0f;
#pragma unroll
        for (int t = 0; t < 4; ++t) {
            v8f cz = {};
            // s(box, point) = (x - lo)(hi - x); >= 0 iff inside on that dim
            v8f s0 = __builtin_amdgcn_wmma_f32_16x16x4_f32(
                false, A0[t], false, B0, (short)0, cz, false, false);
            v8f s1 = __builtin_amdgcn_wmma_f32_16x16x4_f32(
                false, A1[t], false, B1, (short)0, cz, false, false);
#pragma unroll
            for (int j = 0; j < 8; ++j) {
                // inside iff both sign bits clear (all values finite) -> bitwise OR test
                const int u = __float_as_int(s0[j]) | __float_as_int(s1[j]);
                acc += (u >= 0) ? dlt[t][j] : 0.0f;
            }
        }

        // Combine the two lane-halves (boxes j+0..7+16t vs j+8..15+16t): one shuffle.
        acc += __shfl_xor(acc, 16, 32);
        const float r = __logf(S + acc);

        // lanes 0-15 hold points p16+lane: coalesced store
        const int p = p16 + col;
        if (lane < 16 && p < N) out[p] = r;
    }
}

extern "C" void kernel_launch(void* const* d_in, const int* in_sizes, int n_in,
                              void* d_out, int out_size, void* d_ws, size_t ws_size,
                              hipStream_t stream) {
    const float* X         = (const float*)d_in[0];
    const float* weights   = (const float*)d_in[1];
    const float* centers   = (const float*)d_in[2];
    const float* log_sizes = (const float*)d_in[3];
    float* out = (float*)d_out;
    float* ws  = (float*)d_ws;
    const int N = in_sizes[0] / 2;

    umm_setup_kernel<<<1, KBOX, 0, stream>>>(weights, centers, log_sizes, ws);

    const int nStrips = (N + 15) / 16;
    int blocks = (nStrips + 7) / 8;           // 8 waves/block, 1 strip per wave-iteration
    if (blocks > 3072) blocks = 3072;         // grid-stride beyond this
    if (blocks < 1) blocks = 1;
    umm_main_kernel<<<blocks, 256, 0, stream>>>(X, ws, out, N);
}